// AttentionLayer_12919261626817
// MI455X (gfx1250) — compile-verified
//
#include <hip/hip_runtime.h>
#include <math.h>

typedef __attribute__((ext_vector_type(2))) float v2f;
typedef __attribute__((ext_vector_type(8))) float v8f;

// ---------------------------------------------------------------------------
// Batched row-major GEMM  C[M,N] = A[M,K] * B[K,N]  using V_WMMA_F32_16X16X4_F32.
// Register blocking: each wave owns a 32x64 strip (2 m-tiles x 4 n-tiles).
// Per k-step (K advances 4): 2 A-fragment loads (b64) + 8 B loads (b32)
// feed 8 WMMAs -> 1.5 dwords loaded per WMMA, 8 independent acc chains.
// blockDim.x = 256 (8 waves). Grid exactly covers (M/32)*(N/64) wave strips
// so EXEC is all ones at every WMMA. blockIdx.y = batch; strides in elements
// (0 => operand shared across batches).
// ---------------------------------------------------------------------------
__global__ void gemm_f32_wmma(const float* __restrict__ A,
                              const float* __restrict__ B,
                              float* __restrict__ C,
                              int M, int N, int K,
                              long long sA, long long sB, long long sC) {
    const int lane = threadIdx.x & 31;
    const int wave = threadIdx.x >> 5;
    const long long batch = blockIdx.y;

    A += batch * sA;
    B += batch * sB;
    C += batch * sC;

    const int stripsN = N >> 6;                     // 64-wide strips
    const int strip   = blockIdx.x * 8 + wave;      // wave-uniform
    const int sm      = strip / stripsN;            // 32-row strip index
    const int sn      = strip % stripsN;            // 64-col strip index

    const int m16     = lane & 15;
    const int khalf   = (lane >> 4) << 1;           // 0 or 2
    const int row0    = sm * 32 + m16;              // A row, m-tile 0
    const int colBase = sn * 64 + m16;              // B/D column (n-tile 0)

    v8f acc[2][4];
#pragma unroll
    for (int i = 0; i < 2; ++i)
#pragma unroll
        for (int j = 0; j < 4; ++j)
            acc[i][j] = (v8f){};

    const float* Arow0 = A + (long long)row0 * K + khalf;
    const float* Arow1 = Arow0 + (long long)16 * K;

    for (int k = 0; k < K; k += 4) {
        v2f a0 = *(const v2f*)(Arow0 + k);          // m-tile 0: rows sm*32+0..15
        v2f a1 = *(const v2f*)(Arow1 + k);          // m-tile 1: rows sm*32+16..31

        const float* Bp = B + (long long)(k + khalf) * N + colBase;
        v2f bf[4];
#pragma unroll
        for (int j = 0; j < 4; ++j) {
            bf[j].x = Bp[16 * j];                   // K = k+khalf
            bf[j].y = Bp[N + 16 * j];               // K = k+khalf+1
        }

#pragma unroll
        for (int j = 0; j < 4; ++j) {
            acc[0][j] = __builtin_amdgcn_wmma_f32_16x16x4_f32(
                false, a0, false, bf[j], (short)0, acc[0][j], false, false);
            acc[1][j] = __builtin_amdgcn_wmma_f32_16x16x4_f32(
                false, a1, false, bf[j], (short)0, acc[1][j], false, false);
        }
    }

    // C/D layout: vgpr r -> M = tileM + r + (lane>=16 ? 8 : 0), N = colBase + 16*j
    const int mHalf = (lane >> 4) << 3;
#pragma unroll
    for (int i = 0; i < 2; ++i) {
        const int mBase = sm * 32 + i * 16 + mHalf;
#pragma unroll
        for (int r = 0; r < 8; ++r) {
            float* Cr = C + (long long)(mBase + r) * N + colBase;
            Cr[0]  = acc[i][0][r];
            Cr[16] = acc[i][1][r];
            Cr[32] = acc[i][2][r];
            Cr[48] = acc[i][3][r];
        }
    }
}

// ---------------------------------------------------------------------------
// Fused scores + softmax. One block per (b,q); blockDim.x = 256 = Te.
// scores[t] = sum_f tanh(Ws[b,t,f] + Uh[b,q,f]) * v[f]   (TRANS-bound)
// Uh row + V_a staged in LDS; wave w handles t = w, w+8, ... with lanes
// covering f (coalesced Ws reads; Ws[b] = 512 KB stays L2-resident across
// the 256 q-blocks sharing it). Block softmax over Te scores -> e[b,q,:].
// ---------------------------------------------------------------------------
__global__ void scores_softmax_kernel(const float* __restrict__ Ws,  // [B,Te,De]
                                      const float* __restrict__ Uh,  // [B,Td,De]
                                      const float* __restrict__ Va,  // [De]
                                      float* __restrict__ e,         // [B,Td,Te]
                                      int Te, int De) {
    const int b    = blockIdx.y;
    const int q    = blockIdx.x;          // Td = gridDim.x
    const int tid  = threadIdx.x;         // 256 threads
    const int lane = tid & 31;
    const int wave = tid >> 5;

    __shared__ float uh[512];
    __shared__ float vs[512];
    __shared__ float sc[256];
    __shared__ float red[8];

    const float* UhRow = Uh + ((long long)b * gridDim.x + q) * De;
    for (int f = tid; f < De; f += 256) {
        uh[f] = UhRow[f];
        vs[f] = Va[f];
    }
    __syncthreads();

    const float* WsB = Ws + (long long)b * Te * De;
    for (int tt = 0; tt < Te; tt += 8) {
        const int t = tt + wave;
        const float* wr = WsB + (long long)t * De;
        float partial = 0.0f;
        for (int f = lane; f < De; f += 32)
            partial += tanhf(wr[f] + uh[f]) * vs[f];
        for (int off = 16; off > 0; off >>= 1)
            partial += __shfl_xor(partial, off, 32);
        if (lane == 0) sc[t] = partial;
    }
    __syncthreads();

    // ---- softmax over sc[0..Te), one score per thread ----
    const float s = sc[tid];

    float m = s;
    for (int off = 16; off > 0; off >>= 1)
        m = fmaxf(m, __shfl_xor(m, off, 32));
    if (lane == 0) red[wave] = m;
    __syncthreads();
    if (tid == 0) {
        float mm = red[0];
        for (int i = 1; i < 8; ++i) mm = fmaxf(mm, red[i]);
        red[0] = mm;
    }
    __syncthreads();
    m = red[0];
    __syncthreads();   // everyone has read red[0] before it is reused

    const float ex = __expf(s - m);
    float sum = ex;
    for (int off = 16; off > 0; off >>= 1)
        sum += __shfl_xor(sum, off, 32);
    if (lane == 0) red[wave] = sum;
    __syncthreads();
    if (tid == 0) {
        float ss = 0.0f;
        for (int i = 0; i < 8; ++i) ss += red[i];
        red[0] = ss;
    }
    __syncthreads();
    const float inv = 1.0f / red[0];

    e[((long long)b * gridDim.x + q) * Te + tid] = ex * inv;
}

// ---------------------------------------------------------------------------
// Launch: Ws/Uh -> d_ws (8 MB), scores/softmax -> e-slot of d_out,
// context GEMM -> c-slot of d_out.
// ---------------------------------------------------------------------------
extern "C" void kernel_launch(void* const* d_in, const int* in_sizes, int n_in,
                              void* d_out, int out_size, void* d_ws, size_t ws_size,
                              hipStream_t stream) {
    (void)in_sizes; (void)n_in; (void)out_size; (void)ws_size;

    constexpr int B  = 8;
    constexpr int Te = 256;
    constexpr int Td = 256;
    constexpr int De = 512;

    const float* enc = (const float*)d_in[0];   // [B,Te,De]
    const float* dec = (const float*)d_in[1];   // [B,Td,De]
    const float* Wa  = (const float*)d_in[2];   // [De,De]
    const float* Ua  = (const float*)d_in[3];   // [De,De]
    const float* Va  = (const float*)d_in[4];   // [De,1]

    float* c = (float*)d_out;                               // [B,Td,De]
    float* e = c + (size_t)B * Td * De;                     // [B,Td,Te]

    float* Ws = (float*)d_ws;                               // [B,Te,De]
    float* Uh = Ws + (size_t)B * Te * De;                   // [B,Td,De]

    // Phase 1: Ws = enc @ Wa, Uh = dec @ Ua   (flatten batch: M = B*256)
    {
        const int M = B * Te, N = De, K = De;
        const int strips = (M / 32) * (N / 64);             // 512 wave strips
        dim3 grid(strips / 8, 1);                           // 64 blocks
        gemm_f32_wmma<<<grid, 256, 0, stream>>>(enc, Wa, Ws, M, N, K, 0, 0, 0);
        gemm_f32_wmma<<<grid, 256, 0, stream>>>(dec, Ua, Uh, M, N, K, 0, 0, 0);
    }

    // Phase 2: scores + softmax -> e
    {
        dim3 grid(Td, B);
        scores_softmax_kernel<<<grid, 256, 0, stream>>>(Ws, Uh, Va, e, Te, De);
    }

    // Phase 3: c[b] = e[b] @ enc[b]   (M=Td, K=Te, N=De, batched over B)
    {
        const int M = Td, N = De, K = Te;
        const int strips = (M / 32) * (N / 64);             // 64 wave strips
        dim3 grid(strips / 8, B);                           // 8 x 8 blocks
        gemm_f32_wmma<<<grid, 256, 0, stream>>>(e, enc, c, M, N, K,
                                                (long long)Td * Te,
                                                (long long)Te * De,
                                                (long long)Td * De);
    }
}